// VectorQuantizer_37005438222603
// MI455X (gfx1250) — compile-verified
//
#include <hip/hip_runtime.h>

typedef __attribute__((ext_vector_type(2))) float v2f;
typedef __attribute__((ext_vector_type(8))) float v8f;
typedef __attribute__((ext_vector_type(4))) unsigned int u32x4;
typedef __attribute__((ext_vector_type(8))) int i32x8;
typedef __attribute__((ext_vector_type(4))) int i32x4;

#define NTOK   65536      // B*H*W tokens
#define KCB    1024       // codebook entries
#define CCH    64         // channels
#define HW     4096       // H*W
#define CHW    (CCH*HW)
#define KCHUNK 128        // codebook rows staged per LDS chunk
#define NCHUNK (KCB/KCHUNK)
#define LDSS   68         // padded LDS row stride (dwords) -> bank-conflict-free

#if defined(__has_builtin)
#if __has_builtin(__builtin_amdgcn_tensor_load_to_lds) && __has_builtin(__builtin_amdgcn_s_wait_tensorcnt)
#define USE_TDM 1
#endif
#endif

// ---------------------------------------------------------------- init
__global__ void vq_init(float* acc) {
    if (threadIdx.x < 2) acc[threadIdx.x] = 0.0f;
}

// ------------------------------------------------- e2half[k] = 0.5*||e_k||^2
__global__ __launch_bounds__(256) void vq_e2half(const float* __restrict__ emb,
                                                 float* __restrict__ e2half) {
    int w = (blockIdx.x * blockDim.x + threadIdx.x) >> 5;
    int lane = threadIdx.x & 31;
    if (w < KCB) {
        const float* row = emb + (size_t)w * CCH;
        float a = row[lane], b = row[lane + 32];
        float s = a * a + b * b;
        #pragma unroll
        for (int m = 16; m >= 1; m >>= 1) s += __shfl_xor(s, m, 32);
        if (lane == 0) e2half[w] = 0.5f * s;
    }
}

// ------------------------------------------------------------ sum of x^2
__global__ __launch_bounds__(256) void vq_sumx2(const float* __restrict__ x,
                                                float* __restrict__ acc, int n) {
    float s = 0.0f;
    for (int i = blockIdx.x * blockDim.x + threadIdx.x; i < n;
         i += gridDim.x * blockDim.x) {
        float v = x[i];
        s += v * v;
    }
    #pragma unroll
    for (int m = 16; m >= 1; m >>= 1) s += __shfl_xor(s, m, 32);
    if ((threadIdx.x & 31) == 0) atomicAdd(acc, s);
}

#ifdef USE_TDM
// ---- TDM: DMA one 128x64 fp32 codebook chunk into LDS with 4-dword row pad
//      (pad_interval=64 dwords -> LDS row stride 68, matches compute layout)
__device__ __forceinline__ void tdm_load_chunk(const float* gsrc,
                                               unsigned lds_byte_off) {
    unsigned long long ga = (unsigned long long)(uintptr_t)gsrc;
    u32x4 g0;
    g0[0] = 1u;                                            // count=1, user D#
    g0[1] = lds_byte_off;                                  // lds_addr
    g0[2] = (unsigned)ga;                                  // global_addr[31:0]
    g0[3] = (unsigned)((ga >> 32) & 0x01FFFFFFu) | (2u << 30); // addr[56:32]|type=2
    i32x8 g1;
    g1[0] = (int)((2u << 16) | (1u << 20) | (5u << 22) | (3u << 25));
             // data_size=4B | pad_enable | pad_interval=64dw | pad_amount=4dw
    g1[1] = (int)(64u << 16);          // tensor_dim0 = 64 (lo16)
    g1[2] = (int)((unsigned)KCB << 16);// tensor_dim0 hi=0 | tensor_dim1 lo16=1024
    g1[3] = (int)(64u << 16);          // tensor_dim1 hi=0 | tile_dim0 = 64
    g1[4] = (int)KCHUNK;               // tile_dim1 = 128 rows | tile_dim2 = 0
    g1[5] = 64;                        // tensor_dim0_stride lo32 = 64
    g1[6] = 0;                         // stride hi | dim1_stride lo (unused, 2D)
    g1[7] = 0;
    i32x4 z4 = {0, 0, 0, 0};
    i32x8 z8 = {0, 0, 0, 0, 0, 0, 0, 0};
    // 6-arg toolchain form: (g0, g1, g2, g3, <zero>, cpol); g2/g3 zero => 2-D
    __builtin_amdgcn_tensor_load_to_lds(g0, g1, z4, z4, z8, 0);
}
#endif

// ------------------------------------------------- main WMMA argmax kernel
// block = 256 threads = 8 waves; each wave owns 16 tokens; grid = NTOK/128
__global__ __launch_bounds__(256) void vq_main(const float* __restrict__ x,
                                               const float* __restrict__ emb,
                                               const float* __restrict__ e2half,
                                               int*   __restrict__ idxout,
                                               float* __restrict__ fidxout,
                                               float* __restrict__ accS) {
    __shared__ float se[2][KCHUNK * LDSS];       // 2 x 34,816 B (double buffer)

    const int tid   = threadIdx.x;
    const int lane  = tid & 31;
    const int wave  = tid >> 5;
    const int tokenBase = blockIdx.x * 128 + wave * 16;

#ifdef USE_TDM
    if (wave == 0)                               // TDM ignores EXEC: one wave
        tdm_load_chunk(emb, (unsigned)(uintptr_t)&se[0][0]);
#endif

    // ---- A fragments: 16x64 token tile (ISA 32-bit A layout, 16 k-chunks of 4)
    const int m      = lane & 15;
    const int chalf  = (lane >> 4) << 1;         // lanes 0-15: c+0/1, 16-31: c+2/3
    const int n      = tokenBase + m;
    const size_t xbase = (size_t)(n >> 12) * CHW + (size_t)(n & (HW - 1));
    v2f a[16];
    #pragma unroll
    for (int j = 0; j < 16; ++j) {
        const int c0 = 4 * j + chalf;
        a[j].x = x[xbase + (size_t)c0 * HW];
        a[j].y = x[xbase + (size_t)(c0 + 1) * HW];
    }

    float bestv[8];
    int   besti[8];
    #pragma unroll
    for (int r = 0; r < 8; ++r) { bestv[r] = -3.402823466e38f; besti[r] = 0; }

#ifdef USE_TDM
    if (wave == 0) __builtin_amdgcn_s_wait_tensorcnt(0);
    __syncthreads();                             // chunk 0 resident
#endif

    for (int c = 0; c < NCHUNK; ++c) {
        const int buf = c & 1;
        const int k0  = c * KCHUNK;

#ifdef USE_TDM
        if (wave == 0 && c + 1 < NCHUNK)         // overlap DMA of next chunk
            tdm_load_chunk(emb + (size_t)(k0 + KCHUNK) * CCH,
                           (unsigned)(uintptr_t)&se[buf ^ 1][0]);
        const int sbuf = buf;
#else
        {   // ---- synchronous fallback staging (coalesced b128, padded rows)
            const float4* src = (const float4*)(emb + (size_t)k0 * CCH);
            #pragma unroll
            for (int it = 0; it < 8; ++it) {
                int q = tid + it * 256;          // 2048 float4s total
                float4 v = src[q];
                int row = q >> 4, c4 = q & 15;
                *(float4*)&se[0][row * LDSS + c4 * 4] = v;
            }
            __syncthreads();
            if (k0 + KCHUNK < KCB)
                __builtin_prefetch(emb + (size_t)(k0 + KCHUNK) * CCH, 0, 1);
        }
        const int sbuf = 0;
#endif

        #pragma unroll
        for (int t = 0; t < KCHUNK / 16; ++t) {
            // ---- B fragments from LDS (row = codebook entry, padded stride)
            const int    er = t * 16 + (lane & 15);
            const float* ep = &se[sbuf][er * LDSS + chalf];
            v8f s = {};
            #pragma unroll
            for (int j = 0; j < 16; ++j) {       // C=64 => 16 chained k=4 WMMAs
                v2f b;
                b.x = ep[4 * j];
                b.y = ep[4 * j + 1];
                s = __builtin_amdgcn_wmma_f32_16x16x4_f32(
                        false, a[j], false, b, (short)0, s, false, false);
            }
            const int   kc = k0 + t * 16 + (lane & 15);
            const float e2 = e2half[kc];
            #pragma unroll
            for (int r = 0; r < 8; ++r) {        // running argmax of dot - e2/2
                float cand = s[r] - e2;
                if (cand > bestv[r]) { bestv[r] = cand; besti[r] = kc; }
            }
        }

#ifdef USE_TDM
        if (wave == 0) __builtin_amdgcn_s_wait_tensorcnt(0);
#endif
        __syncthreads();
    }

    // ---- per-token reduction: C/D layout => VGPR r holds rows r (lanes 0-15)
    //      and r+8 (lanes 16-31); butterfly within each 16-lane half
    float ssum = 0.0f;
    #pragma unroll
    for (int r = 0; r < 8; ++r) {
        float bv = bestv[r];
        int   bi = besti[r];
        #pragma unroll
        for (int msk = 8; msk >= 1; msk >>= 1) {
            float ov = __shfl_xor(bv, msk, 32);
            int   oi = __shfl_xor(bi, msk, 32);
            if (ov > bv || (ov == bv && oi < bi)) { bv = ov; bi = oi; }
        }
        if ((lane & 15) == 0) {
            const int tok = tokenBase + r + ((lane >> 4) << 3);
            idxout[tok]  = bi;
            fidxout[tok] = (float)bi;
            ssum += bv;                          // s_max = dot - 0.5*||e||^2
        }
    }
    ssum += __shfl_xor(ssum, 16, 32);
    if (lane == 0) atomicAdd(accS, ssum);
}

// ------------------------------------------ x_q[b,c,h,w] = emb[idx[n], c]
__global__ __launch_bounds__(256) void vq_gather(const float* __restrict__ emb,
                                                 const int* __restrict__ idx,
                                                 float* __restrict__ xq) {
    int o = blockIdx.x * blockDim.x + threadIdx.x;   // NTOK*CCH threads
    int b  = o >> 18;                                 // / (C*H*W)
    int c  = (o >> 12) & 63;
    int hw = o & (HW - 1);
    int nn = (b << 12) | hw;
    xq[o] = emb[(size_t)idx[nn] * CCH + c];
}

// --------------------------------- losses = (Sx2 - 2*Ss) / (N*C), both equal
__global__ void vq_finalize(const float* __restrict__ acc,
                            float* __restrict__ losses) {
    if (threadIdx.x == 0) {
        float loss = (acc[1] - 2.0f * acc[0]) * (1.0f / (float)(NTOK * (size_t)CCH));
        losses[0] = loss;   // codebook_loss
        losses[1] = loss;   // commitment_loss
    }
}

extern "C" void kernel_launch(void* const* d_in, const int* in_sizes, int n_in,
                              void* d_out, int out_size, void* d_ws, size_t ws_size,
                              hipStream_t stream) {
    const float* x   = (const float*)d_in[0];   // [16,64,64,64]
    const float* emb = (const float*)d_in[1];   // [1024,64]

    float* xq      = (float*)d_out;             // 4,194,304 floats (NCHW)
    float* losses  = (float*)d_out + 4194304;   // 2 scalars
    float* out_idx = (float*)d_out + 4194306;   // 65,536 indices (as float)

    float* acc    = (float*)d_ws;               // [0]=sum s_max, [1]=sum x^2
    float* e2half = (float*)d_ws + 64;          // 1024 floats
    int*   idxbuf = (int*)((float*)d_ws + 64 + KCB); // 65,536 ints

    vq_init    <<<1,     32,  0, stream>>>(acc);
    vq_e2half  <<<128,   256, 0, stream>>>(emb, e2half);
    vq_sumx2   <<<2048,  256, 0, stream>>>(x, acc + 1, NTOK * CCH);
    vq_main    <<<NTOK / 128, 256, 0, stream>>>(x, emb, e2half, idxbuf, out_idx, acc);
    vq_gather  <<<(NTOK * CCH) / 256, 256, 0, stream>>>(emb, idxbuf, xq);
    vq_finalize<<<1,     32,  0, stream>>>(acc, losses);
}